// LocalAttention2d_80401787781567
// MI455X (gfx1250) — compile-verified
//
#include <hip/hip_runtime.h>
#include <hip/hip_bf16.h>

typedef __attribute__((ext_vector_type(2))) float v2f;
typedef __attribute__((ext_vector_type(8))) float v8f;

#define B_SZ   64
#define QCH    1024
#define HH     64
#define WW     64
#define CTX_SZ 1024
#define RWIN   16
#define CWIN   16
#define WT     256    // RWIN*CWIN
// padded grid is 65x65; index 0 (row or col) is the NaN border

__device__ __forceinline__ float wave_sum(float x) {
#pragma unroll
    for (int o = 16; o > 0; o >>= 1) x += __shfl_xor(x, o, 32);
    return x;
}
__device__ __forceinline__ float wave_max(float x) {
#pragma unroll
    for (int o = 16; o > 0; o >>= 1) x = fmaxf(x, __shfl_xor(x, o, 32));
    return x;
}

// ---------------------------------------------------------------------------
// Kernel 1: p_{x,y}[b] = sigmoid(c_t[b,:] . w_p[k,:]) * 63
// grid = 64 blocks (one per batch), 64 threads (wave0 -> k=0, wave1 -> k=1)
// ---------------------------------------------------------------------------
__global__ void loc_kernel(const float* __restrict__ c_t,
                           const float* __restrict__ w_p,
                           float* __restrict__ loc_ws) {
    const int b = blockIdx.x, t = threadIdx.x;
    const int comp = t >> 5, lane = t & 31;
    const float* ct = c_t + (size_t)b * CTX_SZ;
    const float* wp = w_p + (size_t)comp * CTX_SZ;
    float s = 0.f;
    for (int k = lane; k < CTX_SZ; k += 32) s += ct[k] * wp[k];
    s = wave_sum(s);
    if (lane == 0) loc_ws[b * 2 + comp] = (1.f / (1.f + expf(-s))) * 63.0f;
}

// ---------------------------------------------------------------------------
// Kernel 2: v = c_t (64x1024) @ w_a (1024x1024), fp32 WMMA 16x16x4.
// grid = (1024/64, 64/16) = (16,4), block = 128 (4 waves, each one 16x16 tile)
// A frag (16x4): VGPR0 = K{0|2}, VGPR1 = K{1|3} across lane halves.
// B frag (4x16): VGPR0 = row K{0|2}, VGPR1 = row K{1|3}, N = lane&15.
// C/D (16x16):   VGPR g -> row (g | g+8), N = lane&15.
// ---------------------------------------------------------------------------
__global__ void gemm_v_kernel(const float* __restrict__ c_t,
                              const float* __restrict__ w_a,
                              float* __restrict__ v_ws) {
    const int lane = threadIdx.x & 31;
    const int wv   = threadIdx.x >> 5;
    const int m0   = blockIdx.y * 16;
    const int n0   = blockIdx.x * 64 + wv * 16;
    const int half = (lane < 16) ? 0 : 2;
    const int mrow = m0 + (lane & 15);
    const int ncol = n0 + (lane & 15);

    v8f acc = {0.f, 0.f, 0.f, 0.f, 0.f, 0.f, 0.f, 0.f};
    for (int k = 0; k < CTX_SZ; k += 4) {
        v2f a, bv;
        a.x  = c_t[(size_t)mrow * CTX_SZ + k + half];
        a.y  = c_t[(size_t)mrow * CTX_SZ + k + half + 1];
        bv.x = w_a[(size_t)(k + half) * QCH + ncol];
        bv.y = w_a[(size_t)(k + half + 1) * QCH + ncol];
        acc = __builtin_amdgcn_wmma_f32_16x16x4_f32(false, a, false, bv,
                                                    (short)0, acc, false, false);
    }
    const int rbase = m0 + ((lane < 16) ? 0 : 8);
#pragma unroll
    for (int g = 0; g < 8; ++g)
        v_ws[(size_t)(rbase + g) * QCH + ncol] = acc[g];
}

// ---------------------------------------------------------------------------
// Kernel 3: windowed attention per batch. 64 blocks x 256 threads.
// thread t <-> window element (i=t/16, j=t%16).
// ---------------------------------------------------------------------------
__global__ void attn_kernel(const float* __restrict__ q_i,
                            const float* __restrict__ loc_ws,
                            const float* __restrict__ v_ws,
                            float* __restrict__ out) {
    __shared__ float v_sh[QCH];
    __shared__ float tile_sh[WT * 33];   // stride 33 -> conflict-free
    __shared__ float a_sh[WT];           // partial sums in pass 2
    __shared__ float red_sh[8];
    __shared__ int   r_idx[RWIN];
    __shared__ int   c_idx[CWIN];
    __shared__ float r_gau[RWIN];
    __shared__ float c_gau[CWIN];

    const int b = blockIdx.x, t = threadIdx.x;
    const int lane = t & 31, wv = t >> 5;

    // stage v[b,:] into LDS
    for (int k = t; k < QCH; k += 256) v_sh[k] = v_ws[(size_t)b * QCH + k];

    // window indices + gaussian shifts (clip to [0,65], 65 wraps to NaN border 0)
    if (t < 32) {
        const int comp = t >> 4;          // 0: rows, 1: cols
        const int o = (t & 15) - 7;       // offsets -7..8
        const float p = loc_ws[b * 2 + comp];
        int idx = (int)rintf(p) + o;
        idx = idx < 0 ? 0 : (idx > 65 ? 65 : idx);
        idx = idx % 65;
        const float d = (fmaxf((float)(idx - 1), 0.f) - p) * 0.125f; // /8
        if (comp == 0) { r_idx[t & 15] = idx; r_gau[t & 15] = 2.f * d * d; }
        else           { c_idx[t & 15] = idx; c_gau[t & 15] = 2.f * d * d; }
    }
    __syncthreads();

    const int i = t >> 4, j = t & 15;
    const int r = r_idx[i], c = c_idx[j];
    const bool valid = (r > 0) && (c > 0);
    const float* qbase = q_i + (size_t)b * QCH * HH * WW
                             + (size_t)(r - 1) * WW + (c - 1);

    // ---- pass 1: logits a_t[w] = q_g[w,:] . v  (strided over q, 16KB apart) --
    float acc = 0.f;
    if (valid) {
#pragma unroll 4
        for (int k = 0; k < QCH; ++k) {
            if ((k & 15) == 0) {
                int pk = k + 32 < QCH ? k + 32 : QCH - 1;
                __builtin_prefetch(qbase + (size_t)pk * (HH * WW), 0, 1);
            }
            acc += qbase[(size_t)k * (HH * WW)] * v_sh[k];
        }
    }
    float L = valid ? (acc - (r_gau[i] + c_gau[j])) : -__builtin_inff();

    // ---- softmax over 256 window elements ----
    float m = wave_max(L);
    if (lane == 0) red_sh[wv] = m;
    __syncthreads();
    m = red_sh[0];
#pragma unroll
    for (int x = 1; x < 8; ++x) m = fmaxf(m, red_sh[x]);
    __syncthreads();

    const float e = expf(L - m);          // -inf -> 0
    float s = wave_sum(e);
    if (lane == 0) red_sh[wv] = s;
    __syncthreads();
    s = 0.f;
#pragma unroll
    for (int x = 0; x < 8; ++x) s += red_sh[x];
    const float wt = e / s;
    __syncthreads();

    // ---- pass 2: s_t[q] = sum_w wt[w] * q_g[w,q], via LDS transpose tiles ---
    for (int q0 = 0; q0 < QCH; q0 += 32) {
        if (valid) {
#pragma unroll 4
            for (int qq = 0; qq < 32; ++qq)
                tile_sh[t * 33 + qq] = wt * qbase[(size_t)(q0 + qq) * (HH * WW)];
        } else {
#pragma unroll
            for (int qq = 0; qq < 32; ++qq) tile_sh[t * 33 + qq] = 0.f;
        }
        __syncthreads();
        {
            const int qq = t & 31, g = t >> 5;
            float ps = 0.f;
#pragma unroll 8
            for (int k = 0; k < 32; ++k) ps += tile_sh[(g * 32 + k) * 33 + qq];
            a_sh[(g << 5) + qq] = ps;
        }
        __syncthreads();
        if (t < 32) {
            float tot = 0.f;
#pragma unroll
            for (int g2 = 0; g2 < 8; ++g2) tot += a_sh[(g2 << 5) + t];
            out[(size_t)b * QCH + q0 + t] = tot;
        }
        __syncthreads();
    }
}

extern "C" void kernel_launch(void* const* d_in, const int* in_sizes, int n_in,
                              void* d_out, int out_size, void* d_ws, size_t ws_size,
                              hipStream_t stream) {
    const float* q_i = (const float*)d_in[0];   // (64,1024,64,64)
    const float* c_t = (const float*)d_in[1];   // (64,1024)
    const float* w_a = (const float*)d_in[2];   // (1024,1024)
    const float* w_p = (const float*)d_in[3];   // (2,1024)
    float* out = (float*)d_out;                 // (64,1024)

    float* loc_ws = (float*)d_ws;               // 128 floats (p_x,p_y per batch)
    float* v_ws   = (float*)d_ws + 128;         // 64*1024 floats

    loc_kernel<<<B_SZ, 64, 0, stream>>>(c_t, w_p, loc_ws);

    dim3 g2(QCH / 64, B_SZ / 16);
    gemm_v_kernel<<<g2, 128, 0, stream>>>(c_t, w_a, v_ws);

    attn_kernel<<<B_SZ, 256, 0, stream>>>(q_i, loc_ws, v_ws, out);
}